// ResourceAttentionEmbeddingLayer_71683004171057
// MI455X (gfx1250) — compile-verified
//
#include <hip/hip_runtime.h>

#define HID 8
#define FDIM 128
#define NEG_SLOPE 0.2f

typedef __attribute__((ext_vector_type(2))) float v2f;
typedef __attribute__((ext_vector_type(8))) float v8f;

// ---- monotone float <-> orderable uint for atomicMax on signed floats ----
__device__ __forceinline__ unsigned floatToOrderedUint(float f) {
    unsigned b = __float_as_uint(f);
    return (b & 0x80000000u) ? ~b : (b | 0x80000000u);
}
__device__ __forceinline__ float orderedUintToFloat(unsigned k) {
    unsigned b = (k & 0x80000000u) ? (k & 0x7fffffffu) : ~k;
    return __uint_as_float(b);
}
__device__ __forceinline__ float lrelu(float x) { return (x > 0.f) ? x : NEG_SLOPE * x; }

// =====================================================================
// K0: zero the scatter accumulator + init softmax max/sum cells
// =====================================================================
__global__ __launch_bounds__(256) void init_kernel(float* __restrict__ summed,
                                                   unsigned* __restrict__ gmaxKey,
                                                   float* __restrict__ gsum, int n) {
    int i = blockIdx.x * blockDim.x + threadIdx.x;
    if (i < n) summed[i] = 0.f;
    if (i == 0) { *gmaxKey = 0u; *gsum = 0.f; }
}

// =====================================================================
// K1: Y[nRows x 8] = X[nRows x 128] @ W[128 x 8] via v_wmma_f32_16x16x4_f32
//     One wave per 16-row tile, 8 waves (128 rows) per block.
//     B (128x8) is staged zero-padded to 128x16 in LDS as K-pairs, so the
//     inner loop is: global_load_b64 (A) + ds_load_b64 (B) + v_wmma.
// =====================================================================
__global__ __launch_bounds__(256) void proj_wmma_kernel(const float* __restrict__ X,
                                                        const float* __restrict__ W,
                                                        float* __restrict__ Y, int nRows) {
    // Bp[k2][n] = (W[2*k2][n], W[2*k2+1][n]) for n<8, zeros for n>=8
    __shared__ v2f Bp[(FDIM / 2) * 16];              // 64 x 16 pairs = 8 KB

    for (int idx = threadIdx.x; idx < (FDIM / 2) * 16; idx += 256) {
        const int k2 = idx >> 4;
        const int nn = idx & 15;
        v2f val;
        val.x = (nn < HID) ? W[(2 * k2) * HID + nn]     : 0.f;
        val.y = (nn < HID) ? W[(2 * k2 + 1) * HID + nn] : 0.f;
        Bp[idx] = val;
    }
    __syncthreads();

    const int wave = threadIdx.x >> 5;
    const int lane = threadIdx.x & 31;
    const int tile = blockIdx.x * 8 + wave;
    const int row0 = tile * 16;
    if (row0 >= nRows) return;           // wave-uniform: EXEC stays all-1 for WMMA

    const int m  = lane & 15;            // A row within tile
    const int hi = lane >> 4;            // half-wave selector
    const int n  = m;                    // B/C/D column

    int rowA = row0 + m;
    if (rowA >= nRows) rowA = nRows - 1; // clamp via select (no EXEC divergence)
    const float* __restrict__ arow = X + (size_t)rowA * FDIM;
    const v2f* __restrict__ brow = Bp + hi * 16 + n; // + (k0/2)*16 per step

    v8f acc = {0.f, 0.f, 0.f, 0.f, 0.f, 0.f, 0.f, 0.f};
    #pragma unroll
    for (int k0 = 0; k0 < FDIM; k0 += 4) {
        const int k = k0 + 2 * hi;       // A layout: v0 = K{0|2}, v1 = K{1|3}
        v2f a;
        a.x = arow[k];
        a.y = arow[k + 1];
        const v2f b = brow[(k0 >> 1) * 16];
        acc = __builtin_amdgcn_wmma_f32_16x16x4_f32(false, a, false, b,
                                                    (short)0, acc, false, false);
    }
    // D layout: lane holds D[v + 8*hi][n] in acc[v]
    if (n < HID) {
        #pragma unroll
        for (int v = 0; v < 8; ++v) {
            const int r = row0 + v + 8 * hi;
            if (r < nRows) Y[(size_t)r * HID + n] = acc[v];
        }
    }
}

// =====================================================================
// K2: per-node scalars.
//   selfLogit[i] = leaky_relu(r_i . (a_self[:8] + a_self[8:]))
//   sR[i] = r_i . a_cross[:8],  sO[j] = o_j . a_cross[8:]
//   plus block-max of self logits -> atomicMax(gmaxKey)
// =====================================================================
__global__ __launch_bounds__(256) void node_scalars_kernel(
    const float* __restrict__ r, const float* __restrict__ o,
    const float* __restrict__ a_self, const float* __restrict__ a_cross,
    int R, int O,
    float* __restrict__ selfLogit, float* __restrict__ sR, float* __restrict__ sO,
    unsigned* __restrict__ gmaxKey) {
    int i = blockIdx.x * blockDim.x + threadIdx.x;
    float myLogit = -3.4e38f;
    if (i < R) {
        float d1 = 0.f, d2 = 0.f;
        #pragma unroll
        for (int h = 0; h < HID; ++h) {
            float v = r[(size_t)i * HID + h];
            d1 += v * (a_self[h] + a_self[HID + h]);
            d2 += v * a_cross[h];
        }
        float sl = lrelu(d1);
        selfLogit[i] = sl;
        sR[i] = d2;
        myLogit = sl;
    }
    if (i < O) {
        float d3 = 0.f;
        #pragma unroll
        for (int h = 0; h < HID; ++h)
            d3 += o[(size_t)i * HID + h] * a_cross[HID + h];
        sO[i] = d3;
    }
    __shared__ float sm[256];
    sm[threadIdx.x] = myLogit;
    __syncthreads();
    for (int s = 128; s > 0; s >>= 1) {
        if (threadIdx.x < s) sm[threadIdx.x] = fmaxf(sm[threadIdx.x], sm[threadIdx.x + s]);
        __syncthreads();
    }
    if (threadIdx.x == 0 && sm[0] > -3.0e38f)
        atomicMax(gmaxKey, floatToOrderedUint(sm[0]));
}

// =====================================================================
// K3: edge logits (scalar gathers only) + block max -> atomicMax
// =====================================================================
__global__ __launch_bounds__(256) void edge_logits_kernel(
    const int* __restrict__ edges, int E,
    const float* __restrict__ sR, const float* __restrict__ sO,
    float* __restrict__ crossAtt, unsigned* __restrict__ gmaxKey) {
    int e = blockIdx.x * blockDim.x + threadIdx.x;
    float myLogit = -3.4e38f;
    if (e < E) {
        int src = edges[e];          // edge_ops
        int dst = edges[E + e];      // edge_res
        float l = lrelu(sR[dst] + sO[src]);
        crossAtt[e] = l;
        myLogit = l;
    }
    __shared__ float sm[256];
    sm[threadIdx.x] = myLogit;
    __syncthreads();
    for (int s = 128; s > 0; s >>= 1) {
        if (threadIdx.x < s) sm[threadIdx.x] = fmaxf(sm[threadIdx.x], sm[threadIdx.x + s]);
        __syncthreads();
    }
    if (threadIdx.x == 0 && sm[0] > -3.0e38f)
        atomicMax(gmaxKey, floatToOrderedUint(sm[0]));
}

// =====================================================================
// K4: global sum of exp(x - gmax) over self + cross logits
// =====================================================================
__global__ __launch_bounds__(256) void sumexp_kernel(
    const float* __restrict__ selfLogit, const float* __restrict__ crossAtt,
    int R, int E, const unsigned* __restrict__ gmaxKey, float* __restrict__ gsum) {
    int t = blockIdx.x * blockDim.x + threadIdx.x;
    const float gmax = orderedUintToFloat(*gmaxKey);
    float v = 0.f;
    if (t < R + E) v = expf(((t < R) ? selfLogit[t] : crossAtt[t - R]) - gmax);
    __shared__ float sm[256];
    sm[threadIdx.x] = v;
    __syncthreads();
    for (int s = 128; s > 0; s >>= 1) {
        if (threadIdx.x < s) sm[threadIdx.x] += sm[threadIdx.x + s];
        __syncthreads();
    }
    if (threadIdx.x == 0) atomicAdd(gsum, sm[0]);
}

// =====================================================================
// K5: scatter-add: summed[dst] += softmax(cross) * o[src]  (thread per (edge,h))
// =====================================================================
__global__ __launch_bounds__(256) void scatter_kernel(
    const int* __restrict__ edges, int E,
    const float* __restrict__ crossAtt, const float* __restrict__ o,
    const unsigned* __restrict__ gmaxKey, const float* __restrict__ gsum,
    float* __restrict__ summed) {
    int t = blockIdx.x * blockDim.x + threadIdx.x;
    if (t >= E * HID) return;
    int e = t >> 3, h = t & 7;
    int src = edges[e];
    int dst = edges[E + e];
    const float gmax = orderedUintToFloat(*gmaxKey);
    const float inv  = 1.0f / (*gsum);
    float w = expf(crossAtt[e] - gmax) * inv;
    atomicAdd(&summed[(size_t)dst * HID + h], w * o[(size_t)src * HID + h]);
}

// =====================================================================
// K6: out = elu(softmax(self) * r + summed)
// =====================================================================
__global__ __launch_bounds__(256) void finalize_kernel(
    const float* __restrict__ r, const float* __restrict__ summed,
    const float* __restrict__ selfLogit,
    const unsigned* __restrict__ gmaxKey, const float* __restrict__ gsum,
    float* __restrict__ out, int R) {
    int t = blockIdx.x * blockDim.x + threadIdx.x;
    if (t >= R * HID) return;
    int i = t >> 3;
    const float gmax = orderedUintToFloat(*gmaxKey);
    const float inv  = 1.0f / (*gsum);
    float ns = expf(selfLogit[i] - gmax) * inv;
    float x = ns * r[t] + summed[t];
    out[t] = (x > 0.f) ? x : (expf(x) - 1.f);
}

// =====================================================================
extern "C" void kernel_launch(void* const* d_in, const int* in_sizes, int n_in,
                              void* d_out, int out_size, void* d_ws, size_t ws_size,
                              hipStream_t stream) {
    const float* resources  = (const float*)d_in[0];
    const float* operations = (const float*)d_in[1];
    const int*   edges      = (const int*)d_in[2];   // [2, E] stacked: [ops; res]
    const float* Wr         = (const float*)d_in[3];
    const float* Wo         = (const float*)d_in[4];
    const float* a_self     = (const float*)d_in[5];
    const float* a_cross    = (const float*)d_in[6];
    float* out = (float*)d_out;

    const int R = in_sizes[0] / FDIM;
    const int O = in_sizes[1] / FDIM;
    const int E = in_sizes[2] / 2;

    // workspace layout (floats)
    float* ws        = (float*)d_ws;
    float* r         = ws;                          // R*8
    float* o         = r + (size_t)R * HID;         // O*8
    float* selfLogit = o + (size_t)O * HID;         // R
    float* sR        = selfLogit + R;               // R
    float* sO        = sR + R;                      // O
    float* crossAtt  = sO + O;                      // E
    float* summed    = crossAtt + E;                // R*8
    unsigned* gmaxKey = (unsigned*)(summed + (size_t)R * HID);
    float*    gsum    = (float*)(gmaxKey + 1);

    const int BLK = 256;

    // K0: init accumulators
    {
        int n = R * HID;
        init_kernel<<<(n + BLK - 1) / BLK, BLK, 0, stream>>>(summed, gmaxKey, gsum, n);
    }
    // K1: projections via WMMA (one wave per 16-row tile, 8 waves/block)
    {
        int tilesR = (R + 15) / 16;
        proj_wmma_kernel<<<(tilesR + 7) / 8, BLK, 0, stream>>>(resources, Wr, r, R);
        int tilesO = (O + 15) / 16;
        proj_wmma_kernel<<<(tilesO + 7) / 8, BLK, 0, stream>>>(operations, Wo, o, O);
    }
    // K2: per-node scalars + self-logit max
    {
        int n = (R > O) ? R : O;
        node_scalars_kernel<<<(n + BLK - 1) / BLK, BLK, 0, stream>>>(
            r, o, a_self, a_cross, R, O, selfLogit, sR, sO, gmaxKey);
    }
    // K3: edge logits + max
    edge_logits_kernel<<<(E + BLK - 1) / BLK, BLK, 0, stream>>>(edges, E, sR, sO,
                                                               crossAtt, gmaxKey);
    // K4: global sum of exp
    {
        int n = R + E;
        sumexp_kernel<<<(n + BLK - 1) / BLK, BLK, 0, stream>>>(selfLogit, crossAtt,
                                                               R, E, gmaxKey, gsum);
    }
    // K5: scatter-add weighted ops features
    {
        int n = E * HID;
        scatter_kernel<<<(n + BLK - 1) / BLK, BLK, 0, stream>>>(edges, E, crossAtt, o,
                                                                gmaxKey, gsum, summed);
    }
    // K6: ELU epilogue
    {
        int n = R * HID;
        finalize_kernel<<<(n + BLK - 1) / BLK, BLK, 0, stream>>>(r, summed, selfLogit,
                                                                 gmaxKey, gsum, out, R);
    }
}